// LatentCorrelationLayer_33543694581976
// MI455X (gfx1250) — compile-verified
//
#include <hip/hip_runtime.h>
#include <math.h>

// Problem sizes (fixed by the reference)
#define UN 1024          // UNIT = hidden size = node count
#define BB 32            // BATCH
#define TT 128           // TSTEP = GRU input size
#define GG 3072          // 3*UNIT (gate dim)
#define NN2 (UN*UN)      // 1024*1024

#define NWG 64           // persistent workgroups for the GRU scan
#define GRU_THREADS 192  // 6 waves: 2 batch tiles x 3 gates
#define KSLAB 128        // K staging slab for the recurrent GEMM

typedef float v2f __attribute__((ext_vector_type(2)));
typedef float v8f __attribute__((ext_vector_type(8)));
typedef unsigned int u32x4 __attribute__((ext_vector_type(4)));
typedef int i32x4 __attribute__((ext_vector_type(4)));
typedef int i32x8 __attribute__((ext_vector_type(8)));

#ifndef __has_builtin
#define __has_builtin(x) 0
#endif
#if __has_builtin(__builtin_amdgcn_tensor_load_to_lds)
#define HAVE_TDM 1
#else
#define HAVE_TDM 0
#endif

// D = A(16x4 f32) * B(4x16 f32) + C(16x16 f32), full f32 precision.
static __device__ __forceinline__ v8f wmma4(v2f a, v2f b, v8f c) {
  return __builtin_amdgcn_wmma_f32_16x16x4_f32(
      /*neg_a=*/false, a, /*neg_b=*/false, b,
      /*c_mod=*/(short)0, c, /*reuse_a=*/false, /*reuse_b=*/false);
}

#if HAVE_TDM
// Issue one 2-D TDM tile load: rows of tile_d0 f32 elements, tile_d1 rows,
// row stride stride0 (elements), packed contiguously at LDS offset lds_off.
// D# layout per CDNA5 ISA ch.8 (group0: count/lds/global/type; group1: dims).
// This toolchain's builtin is the 6-arg form:
//   (u32x4 g0, i32x8 g1, i32x4 g2, i32x4 g3, i32x8 g4, i32 cpol)
static __device__ __forceinline__ void tdm_load_2d(
    unsigned lds_off, const float* gptr,
    unsigned tensor_d0, unsigned tensor_d1,
    unsigned tile_d0, unsigned tile_d1, unsigned stride0) {
  unsigned long long ga = (unsigned long long)gptr;
  u32x4 g0;
  g0.x = 1u;                                   // count=1, user descriptor
  g0.y = lds_off;                              // LDS byte address
  g0.z = (unsigned)ga;                         // global_addr[31:0]
  g0.w = ((unsigned)(ga >> 32) & 0x01FFFFFFu)  // global_addr[56:32]
         | (2u << 30);                         // type = 2 ("image")
  i32x8 g1;
  g1[0] = (int)(2u << 16);                     // data_size = 2 -> 4 bytes
  g1[1] = (int)(tensor_d0 << 16);              // tensor_dim0[15:0] @ bits63:48
  g1[2] = (int)((tensor_d0 >> 16) | (tensor_d1 << 16));   // dim0 hi | dim1 lo
  g1[3] = (int)((tensor_d1 >> 16) | (tile_d0 << 16));     // dim1 hi | tile0
  g1[4] = (int)(tile_d1 & 0xFFFFu);            // tile_dim1 (tile_dim2 = 0)
  g1[5] = (int)stride0;                        // tensor_dim0_stride[31:0]
  g1[6] = 0;                                   // stride0 hi | stride1 lo
  g1[7] = 0;
  i32x4 gz4 = {0, 0, 0, 0};                    // groups 2/3 unused (2-D tile)
  i32x8 gz8 = {0, 0, 0, 0, 0, 0, 0, 0};
  __builtin_amdgcn_tensor_load_to_lds(g0, g1, gz4, gz4, gz8, 0);
}
#endif

// ---------------------------------------------------------------- utilities
__global__ void zero_kernel(float* p, int n) {
  int i = blockIdx.x * 256 + threadIdx.x;
  if (i < n) p[i] = 0.0f;
}

__global__ void transpose_whh(const float* __restrict__ whh, float* __restrict__ whhT) {
  int idx = blockIdx.x * 256 + threadIdx.x;      // over UN*GG, output-indexed
  if (idx >= UN * GG) return;
  int k = idx / GG, g = idx - k * GG;
  whhT[idx] = whh[g * UN + k];                   // whhT[k][g] = w_hh[g][k]
}

__global__ void transpose_wih(const float* __restrict__ wih, float* __restrict__ wihT) {
  int idx = blockIdx.x * 256 + threadIdx.x;      // over TT*GG
  if (idx >= TT * GG) return;
  int t = idx / GG, g = idx - t * GG;
  wihT[idx] = wih[g * TT + t];                   // wihT[t][g] = w_ih[g][t]
}

// --------------------------------------------------------- grid-wide barrier
static __device__ __forceinline__ void grid_barrier(unsigned* cnt, unsigned* gen) {
  __syncthreads();
  if (threadIdx.x == 0) {
    unsigned g = __hip_atomic_load(gen, __ATOMIC_RELAXED, __HIP_MEMORY_SCOPE_AGENT);
    __threadfence();
    unsigned prev = __hip_atomic_fetch_add(cnt, 1u, __ATOMIC_ACQ_REL, __HIP_MEMORY_SCOPE_AGENT);
    if (prev == NWG - 1) {
      __hip_atomic_store(cnt, 0u, __ATOMIC_RELAXED, __HIP_MEMORY_SCOPE_AGENT);
      __hip_atomic_fetch_add(gen, 1u, __ATOMIC_RELEASE, __HIP_MEMORY_SCOPE_AGENT);
    } else {
      while (__hip_atomic_load(gen, __ATOMIC_ACQUIRE, __HIP_MEMORY_SCOPE_AGENT) == g) {
        __builtin_amdgcn_s_sleep(2);
      }
    }
  }
  __syncthreads();
}

// ------------------------------------------------------- persistent GRU scan
// 64 WGs, each owns 16 hidden columns. 6 waves per WG: wave w -> batch tile
// (w&1), gate (w>>1) in {r,z,n}. f32 WMMA; h double-buffered in global;
// TDM streams the xs gather + double-buffered h K-slabs into LDS while the
// waves run WMMA (sync via s_wait_tensorcnt); grid barrier between steps.
__global__ __launch_bounds__(GRU_THREADS)
void gru_kernel(const float* __restrict__ x,
                const float* __restrict__ b_ih, const float* __restrict__ b_hh,
                const float* __restrict__ wkey, const float* __restrict__ wqry,
                const float* __restrict__ whhT, const float* __restrict__ wihT,
                float* __restrict__ hbuf,
                float* __restrict__ keyv, float* __restrict__ qryv,
                unsigned* __restrict__ bar) {
  __shared__ float x_lds[BB * TT];           // 16 KB: xs[s] slab  [b][t]
  __shared__ float h_lds[2][BB * KSLAB];     // 32 KB: double-buffered h slab
  __shared__ float gh_lds[3][BB][16];        // 6 KB : h-gate tiles
  __shared__ float gi_lds[3][BB][16];        // 6 KB : i-gate tiles

  const int tid  = threadIdx.x;
  const int lane = tid & 31;
  const int w    = tid >> 5;              // 0..5
  const int mt   = w & 1;                 // batch tile
  const int gt   = w >> 1;                // gate: 0=r 1=z 2=n
  const int ln   = lane & 15;
  const int hi   = lane >> 4;             // 0/1 (K phase for f32 WMMA layout)
  const int jcol0 = blockIdx.x << 4;      // 16 hidden columns owned by this WG
  const int m0   = mt << 4;
  const int g0   = gt * UN + jcol0;       // column offset in gate space

  unsigned* cnt = bar;
  unsigned* gen = bar + 1;

#if HAVE_TDM
  const unsigned x_off  = (unsigned)(size_t)&x_lds[0];      // LDS byte addrs
  const unsigned h_off0 = (unsigned)(size_t)&h_lds[0][0];
  const unsigned h_off1 = (unsigned)(size_t)&h_lds[1][0];
#endif

  for (int s = 0; s < UN; ++s) {
    const float* hprev = hbuf + (size_t)(s & 1) * BB * UN;
    float*       hnext = hbuf + (size_t)((s & 1) ^ 1) * BB * UN;

#if HAVE_TDM
    if (w == 0) {
      // xs[s] gather: 4096 rows of 1 element, row stride UN -> packed [b][t]
      tdm_load_2d(x_off, x + s, 1u, (unsigned)(BB * TT),
                  1u, (unsigned)(BB * TT), (unsigned)UN);
      // first h K-slab: 32 rows x 128 elems out of the [32][1024] h matrix
      tdm_load_2d(h_off0, hprev, (unsigned)UN, (unsigned)BB,
                  (unsigned)KSLAB, (unsigned)BB, (unsigned)UN);
      __builtin_amdgcn_s_wait_tensorcnt(1);  // oldest (x) landed; h0 in flight
    }
    __syncthreads();
#else
    for (int idx = tid; idx < BB * TT; idx += GRU_THREADS) {
      int b = idx >> 7, t = idx & (TT - 1);
      x_lds[idx] = x[((size_t)b * TT + t) * UN + s];
    }
    __syncthreads();
#endif

    // input-gate GEMM: ii[b][g] = sum_t xs[b][t] * w_ih[g][t]   (K = 128)
    v8f ci = {0.f, 0.f, 0.f, 0.f, 0.f, 0.f, 0.f, 0.f};
    {
      const float* arow = &x_lds[(m0 + ln) * TT];
      for (int t = 0; t < TT; t += 4) {
        v2f a, bv;
        a.x  = arow[t + 2 * hi];
        a.y  = arow[t + 2 * hi + 1];
        bv.x = wihT[(size_t)(t + 2 * hi) * GG + g0 + ln];
        bv.y = wihT[(size_t)(t + 2 * hi + 1) * GG + g0 + ln];
        ci = wmma4(a, bv, ci);
      }
    }

    // recurrent GEMM: gh[b][g] = sum_k h[b][k] * w_hh[g][k]     (K = 1024)
    v8f ch = {0.f, 0.f, 0.f, 0.f, 0.f, 0.f, 0.f, 0.f};
    int pb = 0;
    for (int k0 = 0; k0 < UN; k0 += KSLAB) {
#if HAVE_TDM
      if (w == 0) {
        if (k0 + KSLAB < UN) {
          // prefetch next slab into the other buffer via TDM
          tdm_load_2d(pb ? h_off0 : h_off1, hprev + k0 + KSLAB,
                      (unsigned)UN, (unsigned)BB,
                      (unsigned)KSLAB, (unsigned)BB, (unsigned)UN);
          __builtin_amdgcn_s_wait_tensorcnt(1);   // slab k0 complete
        } else {
          __builtin_amdgcn_s_wait_tensorcnt(0);   // last slab complete
        }
      }
      __syncthreads();                            // slab pb visible to all
#else
      __syncthreads();
      for (int idx = tid; idx < BB * KSLAB; idx += GRU_THREADS) {
        int b = idx >> 7, kk = idx & (KSLAB - 1);
        h_lds[pb][idx] = hprev[(size_t)b * UN + k0 + kk];
      }
      __syncthreads();
#endif
      if (k0 + KSLAB < UN)   // warm the L2->L0 path for the next weight slab
        __builtin_prefetch(&whhT[(size_t)(k0 + KSLAB) * GG + g0 + ln], 0, 1);
      const float* arow = &h_lds[pb][(m0 + ln) * KSLAB];
      for (int kk = 0; kk < KSLAB; kk += 4) {
        v2f a, bv;
        a.x  = arow[kk + 2 * hi];
        a.y  = arow[kk + 2 * hi + 1];
        bv.x = whhT[(size_t)(k0 + kk + 2 * hi) * GG + g0 + ln];
        bv.y = whhT[(size_t)(k0 + kk + 2 * hi + 1) * GG + g0 + ln];
        ch = wmma4(a, bv, ch);
      }
      __syncthreads();       // done reading slab pb before TDM refills it
      pb ^= 1;
    }

    // scatter WMMA C tiles (lane L, vgpr v -> row v+8*(L/16), col L%16)
    for (int v = 0; v < 8; ++v) {
      int row = m0 + v + 8 * hi;
      gh_lds[gt][row][ln] = ch[v];
      gi_lds[gt][row][ln] = ci[v];
    }
    __syncthreads();

    // pointwise GRU cell + fused attention key/query projections
    for (int idx = tid; idx < BB * 16; idx += GRU_THREADS) {
      int b = idx >> 4, jj = idx & 15;
      int j = jcol0 + jj;
      float ir  = gi_lds[0][b][jj] + b_ih[j];
      float iz  = gi_lds[1][b][jj] + b_ih[UN + j];
      float in_ = gi_lds[2][b][jj] + b_ih[2 * UN + j];
      float hr  = gh_lds[0][b][jj] + b_hh[j];
      float hz  = gh_lds[1][b][jj] + b_hh[UN + j];
      float hn  = gh_lds[2][b][jj] + b_hh[2 * UN + j];
      float r = 1.0f / (1.0f + expf(-(ir + hr)));
      float z = 1.0f / (1.0f + expf(-(iz + hz)));
      float nv = tanhf(in_ + r * hn);
      float hp = hprev[(size_t)b * UN + j];
      float hnew = (1.0f - z) * nv + z * hp;
      hnext[(size_t)b * UN + j] = hnew;
      // key_v[b,s] = dot(out[s,b,:], wkey); accumulate this WG's 16 columns
      atomicAdd(&keyv[(size_t)b * UN + s], hnew * wkey[j]);
      atomicAdd(&qryv[(size_t)b * UN + s], hnew * wqry[j]);
    }
    grid_barrier(cnt, gen);
  }
}

// -------------------------------------------------------------- attention
__global__ void qmax_kernel(const float* __restrict__ qryv, float* __restrict__ qmax) {
  int b = blockIdx.x, tid = threadIdx.x;
  __shared__ float red[256];
  float m = -1e30f;
  for (int j = tid; j < UN; j += 256) m = fmaxf(m, qryv[(size_t)b * UN + j]);
  red[tid] = m; __syncthreads();
  for (int off = 128; off > 0; off >>= 1) {
    if (tid < off) red[tid] = fmaxf(red[tid], red[tid + off]);
    __syncthreads();
  }
  if (tid == 0) qmax[b] = red[0];
}

// one block per row i: mean-over-batch softmax + row degree
__global__ void att_kernel(const float* __restrict__ keyv, const float* __restrict__ qryv,
                           const float* __restrict__ qmax,
                           float* __restrict__ attp, float* __restrict__ degree) {
  int i = blockIdx.x, tid = threadIdx.x;
  __shared__ float red[256];
  float acc[4] = {0.f, 0.f, 0.f, 0.f};
  for (int b = 0; b < BB; ++b) {
    float kb = keyv[(size_t)b * UN + i];
    float mlog = kb + qmax[b];                       // LeakyReLU is monotone:
    mlog = mlog > 0.f ? mlog : 0.2f * mlog;          // row max = leaky(k+qmax)
    float e[4]; float psum = 0.f;
    for (int c = 0; c < 4; ++c) {
      float l = kb + qryv[(size_t)b * UN + c * 256 + tid];
      l = l > 0.f ? l : 0.2f * l;
      float ev = expf(l - mlog);
      e[c] = ev; psum += ev;
    }
    red[tid] = psum; __syncthreads();
    for (int off = 128; off > 0; off >>= 1) {
      if (tid < off) red[tid] += red[tid + off];
      __syncthreads();
    }
    float inv = 1.0f / red[0];
    __syncthreads();
    for (int c = 0; c < 4; ++c) acc[c] += e[c] * inv;
  }
  float dsum = 0.f;
  for (int c = 0; c < 4; ++c) {
    float a = acc[c] * (1.0f / BB);
    attp[(size_t)i * UN + c * 256 + tid] = a;
    dsum += a;
  }
  red[tid] = dsum; __syncthreads();
  for (int off = 128; off > 0; off >>= 1) {
    if (tid < off) red[tid] += red[tid + off];
    __syncthreads();
  }
  if (tid == 0) degree[i] = red[0];
}

// symmetrize + Laplacian; emits attention output, L0 (=0) and L1 (=lap)
__global__ void lap_kernel(const float* __restrict__ attp, const float* __restrict__ degree,
                           float* __restrict__ out_att, float* __restrict__ out_L0,
                           float* __restrict__ out_L1, float* __restrict__ lap) {
  int idx = blockIdx.x * 256 + threadIdx.x;
  if (idx >= NN2) return;
  int i = idx >> 10, j = idx & (UN - 1);
  float sym = 0.5f * (attp[idx] + attp[(size_t)j * UN + i]);
  out_att[idx] = sym;
  float di = degree[i], dj = degree[j];
  float dvi = 1.0f / (sqrtf(di) + 1e-7f);
  float dvj = 1.0f / (sqrtf(dj) + 1e-7f);
  float lv = dvi * ((i == j ? di : 0.0f) - sym) * dvj;
  lap[idx] = lv;
  out_L1[idx] = lv;
  out_L0[idx] = 0.0f;
}

// ---------------------------------------------- Chebyshev GEMM: D = 2*A@B + beta*C
__global__ __launch_bounds__(256)
void cheb_kernel(const float* __restrict__ A, const float* __restrict__ Bm,
                 const float* __restrict__ Cm, float beta, float* __restrict__ D) {
  int w = threadIdx.x >> 5;
  int lane = threadIdx.x & 31;
  int ln = lane & 15, hi = lane >> 4;
  int tile = blockIdx.x * 8 + w;          // 4096 tiles of 16x16, 8 waves/block
  int i0 = (tile >> 6) << 4;
  int j0 = (tile & 63) << 4;
  v8f c = {0.f, 0.f, 0.f, 0.f, 0.f, 0.f, 0.f, 0.f};
  for (int k = 0; k < UN; k += 4) {
    v2f a, bv;
    a.x  = A[(size_t)(i0 + ln) * UN + k + 2 * hi];
    a.y  = A[(size_t)(i0 + ln) * UN + k + 2 * hi + 1];
    bv.x = Bm[(size_t)(k + 2 * hi) * UN + j0 + ln];
    bv.y = Bm[(size_t)(k + 2 * hi + 1) * UN + j0 + ln];
    c = wmma4(a, bv, c);
  }
  for (int v = 0; v < 8; ++v) {
    int row = i0 + v + 8 * hi;
    int col = j0 + ln;
    float extra = Cm ? beta * Cm[(size_t)row * UN + col] : 0.0f;
    D[(size_t)row * UN + col] = 2.0f * c[v] + extra;
  }
}

// ------------------------------------------------------------------ launch
extern "C" void kernel_launch(void* const* d_in, const int* in_sizes, int n_in,
                              void* d_out, int out_size, void* d_ws, size_t ws_size,
                              hipStream_t stream) {
  const float* x    = (const float*)d_in[0];
  const float* w_ih = (const float*)d_in[1];
  const float* w_hh = (const float*)d_in[2];
  const float* b_ih = (const float*)d_in[3];
  const float* b_hh = (const float*)d_in[4];
  const float* wkey = (const float*)d_in[5];
  const float* wqry = (const float*)d_in[6];
  float* out = (float*)d_out;   // [L0 | L1 | L2 | L3 | attention], NN2 each

  float* ws = (float*)d_ws;     // f32 workspace layout (element offsets)
  size_t o_whhT = 0;                                  // [1024][3072]
  size_t o_wihT = o_whhT + (size_t)UN * GG;           // [128][3072]
  size_t o_h    = o_wihT + (size_t)TT * GG;           // 2 x [32][1024]
  size_t o_key  = o_h    + (size_t)2 * BB * UN;       // [32][1024]
  size_t o_qry  = o_key  + (size_t)BB * UN;           // [32][1024]
  size_t o_bar  = o_qry  + (size_t)BB * UN;           // 2 x u32 (padded to 4)
  size_t o_qmax = o_bar  + 4;                         // [32]
  size_t o_deg  = o_qmax + BB;                        // [1024]
  size_t o_att  = o_deg  + UN;                        // [1024][1024] pre-sym
  size_t o_lap  = o_att  + (size_t)NN2;               // [1024][1024]

  unsigned* bar = (unsigned*)(ws + o_bar);

  // zero h double-buffer, key/query accumulators, and barrier words
  int zn = 2 * BB * UN + BB * UN + BB * UN + 4;
  zero_kernel<<<(zn + 255) / 256, 256, 0, stream>>>(ws + o_h, zn);

  transpose_whh<<<(UN * GG + 255) / 256, 256, 0, stream>>>(w_hh, ws + o_whhT);
  transpose_wih<<<(TT * GG + 255) / 256, 256, 0, stream>>>(w_ih, ws + o_wihT);

  gru_kernel<<<NWG, GRU_THREADS, 0, stream>>>(
      x, b_ih, b_hh, wkey, wqry, ws + o_whhT, ws + o_wihT,
      ws + o_h, ws + o_key, ws + o_qry, bar);

  qmax_kernel<<<BB, 256, 0, stream>>>(ws + o_qry, ws + o_qmax);
  att_kernel<<<UN, 256, 0, stream>>>(ws + o_key, ws + o_qry, ws + o_qmax,
                                     ws + o_att, ws + o_deg);
  lap_kernel<<<(NN2 + 255) / 256, 256, 0, stream>>>(
      ws + o_att, ws + o_deg,
      out + (size_t)4 * NN2,   // attention
      out,                     // L0
      out + (size_t)NN2,       // L1
      ws + o_lap);

  // L2 = 2*lap@lap ; L3 = 2*lap@L2 - lap
  cheb_kernel<<<512, 256, 0, stream>>>(ws + o_lap, ws + o_lap, nullptr, 0.0f,
                                       out + (size_t)2 * NN2);
  cheb_kernel<<<512, 256, 0, stream>>>(ws + o_lap, out + (size_t)2 * NN2,
                                       ws + o_lap, -1.0f, out + (size_t)3 * NN2);
}